// NearestNeighborMLP_34772055228495
// MI455X (gfx1250) — compile-verified
//
#include <hip/hip_runtime.h>
#include <hip/hip_bf16.h>
#include <math.h>

#define BATCH  8
#define NTRK   1024
#define NKNN   4
#define EMBD   8
#define WAVES  4            // waves per block, each scans NTRK/WAVES columns
#define CTILES (NTRK / 16 / WAVES)   // 16 column tiles (of 16) per wave

typedef float v2f __attribute__((ext_vector_type(2)));
typedef float v8f __attribute__((ext_vector_type(8)));

// Branchless compare-swap bubble insert of (c, cj) into ascending sorted top-4.
// Strict '<' keeps earliest-seen entry on exact ties (per-lane candidate stream
// is ascending in j -> matches jax.lax.top_k index-stable tie behaviour).
__device__ __forceinline__ void ins4(float c, int cj,
                                     float& d0, float& d1, float& d2, float& d3,
                                     int& j0, int& j1, int& j2, int& j3) {
    bool s0 = c < d0;
    float t0 = s0 ? d0 : c;   int t0j = s0 ? j0 : cj;
    d0 = s0 ? c : d0;         j0 = s0 ? cj : j0;
    bool s1 = t0 < d1;
    float t1 = s1 ? d1 : t0;  int t1j = s1 ? j1 : t0j;
    d1 = s1 ? t0 : d1;        j1 = s1 ? t0j : j1;
    bool s2 = t1 < d2;
    float t2 = s2 ? d2 : t1;  int t2j = s2 ? j2 : t1j;
    d2 = s2 ? t1 : d2;        j2 = s2 ? t1j : j2;
    bool s3 = t2 < d3;
    d3 = s3 ? t2 : d3;        j3 = s3 ? t2j : j3;
}

// Tie-breaking variant for cross-lane / cross-wave merges: equal key -> smaller j wins.
__device__ __forceinline__ void ins4_tb(float c, int cj,
                                        float& d0, float& d1, float& d2, float& d3,
                                        int& j0, int& j1, int& j2, int& j3) {
    bool s0 = (c < d0) || (c == d0 && cj < j0);
    float t0 = s0 ? d0 : c;   int t0j = s0 ? j0 : cj;
    d0 = s0 ? c : d0;         j0 = s0 ? cj : j0;
    bool s1 = (t0 < d1) || (t0 == d1 && t0j < j1);
    float t1 = s1 ? d1 : t0;  int t1j = s1 ? j1 : t0j;
    d1 = s1 ? t0 : d1;        j1 = s1 ? t0j : j1;
    bool s2 = (t1 < d2) || (t1 == d2 && t1j < j2);
    float t2 = s2 ? d2 : t1;  int t2j = s2 ? j2 : t1j;
    d2 = s2 ? t1 : d2;        j2 = s2 ? t1j : j2;
    bool s3 = (t2 < d3) || (t2 == d3 && t2j < j3);
    d3 = s3 ? t2 : d3;        j3 = s3 ? t2j : j3;
}

// Block = 4 waves, handles (batch b, 16 query rows i0..i0+15).
// Wave w scans neighbour columns [w*256, w*256+256).
// WMMA key[j,i] = |p_j|^2 - 2 p_i.p_j  (rank-equivalent to distance for fixed i):
//   A row j = [x_j, y_j, |p_j|^2, 0],  B col i = [-2x_i, -2y_i, 1, 0].
__global__ __launch_bounds__(32 * WAVES)
void knn_mlp_wmma_kernel(const float* __restrict__ obs1,
                         const float* __restrict__ obs2,
                         const float* __restrict__ W,
                         const float* __restrict__ bias,
                         float* __restrict__ out) {
    const int tid  = threadIdx.x;
    const int lane = tid & 31;
    const int wave = tid >> 5;
    const int b    = blockIdx.x >> 6;        // batch
    const int i0   = (blockIdx.x & 63) << 4; // query row tile
    const int half = lane >> 4;              // 0: M 0..7 / K0,K1  1: M 8..15 / K2,K3
    const int l16  = lane & 15;
    const int base = b * NTRK;

    const float2* p2 = (const float2*)obs2;
    const float2* p1 = (const float2*)obs1;

    __shared__ float sd[WAVES][16][NKNN];
    __shared__ int   sj[WAVES][16][NKNN];

    // prefetch this wave's 2KB column slice into cache (global_prefetch)
    const int jt0 = wave * CTILES;           // first column tile for this wave
    __builtin_prefetch((const char*)(p2 + base + jt0 * 16) + lane * 64, 0, 0);

    // ---- B matrix (queries i0..i0+15), loop-invariant ----
    const float2 pi = p2[base + i0 + l16];
    v2f Bm;
    Bm.x = half ? 1.0f : (-2.0f * pi.x);
    Bm.y = half ? 0.0f : (-2.0f * pi.y);

    const int irow = i0 + l16;
    float d0 = INFINITY, d1 = INFINITY, d2 = INFINITY, d3 = INFINITY;
    int   j0 = 0, j1 = 0, j2 = 0, j3 = 0;

    // software-pipelined column-tile loop: load(t+1) overlaps insert-chain(t)
    float2 pj = p2[base + jt0 * 16 + l16];
    for (int t = 0; t < CTILES; ++t) {
        const int tn = (t + 1 < CTILES) ? (t + 1) : t;
        const float2 pjn = p2[base + (jt0 + tn) * 16 + l16];

        const int jt = jt0 + t;
        const float sq = pj.x * pj.x + pj.y * pj.y;
        v2f Am;
        Am.x = half ? sq   : pj.x;   // lanes16-31: K2 = |p_j|^2
        Am.y = half ? 0.0f : pj.y;   // lanes16-31: K3 = 0

        v8f acc = {};
        acc = __builtin_amdgcn_wmma_f32_16x16x4_f32(
            false, Am, false, Bm, (short)0, acc, false, false);

        const int jb = jt * 16 + half * 8;   // this lane's j for acc[v] = jb+v
#pragma unroll
        for (int v = 0; v < 8; ++v) {
            float c  = acc[v];
            int   cj = jb + v;
            c = (cj == irow) ? INFINITY : c;          // exclude diagonal
            if (__ballot(c < d3)) {                   // wave-uniform early-out
                ins4(c, cj, d0, d1, d2, d3, j0, j1, j2, j3);
            }
        }
        pj = pjn;
    }

    // ---- merge lane pair {l, l+16} (columns split across halves) ----
    float pd0 = __shfl_xor(d0, 16), pd1 = __shfl_xor(d1, 16),
          pd2 = __shfl_xor(d2, 16), pd3 = __shfl_xor(d3, 16);
    int   pq0 = __shfl_xor(j0, 16), pq1 = __shfl_xor(j1, 16),
          pq2 = __shfl_xor(j2, 16), pq3 = __shfl_xor(j3, 16);
    ins4_tb(pd0, pq0, d0, d1, d2, d3, j0, j1, j2, j3);
    ins4_tb(pd1, pq1, d0, d1, d2, d3, j0, j1, j2, j3);
    ins4_tb(pd2, pq2, d0, d1, d2, d3, j0, j1, j2, j3);
    ins4_tb(pd3, pq3, d0, d1, d2, d3, j0, j1, j2, j3);

    // ---- cross-wave merge via LDS: waves 1..3 publish, wave 0 reduces ----
    if (wave != 0 && half == 0) {
        sd[wave][l16][0] = d0;  sj[wave][l16][0] = j0;
        sd[wave][l16][1] = d1;  sj[wave][l16][1] = j1;
        sd[wave][l16][2] = d2;  sj[wave][l16][2] = j2;
        sd[wave][l16][3] = d3;  sj[wave][l16][3] = j3;
    }
    __syncthreads();

    if (wave == 0 && half == 0) {
#pragma unroll
        for (int w = 1; w < WAVES; ++w) {
#pragma unroll
            for (int k = 0; k < NKNN; ++k) {
                ins4_tb(sd[w][l16][k], sj[w][l16][k],
                        d0, d1, d2, d3, j0, j1, j2, j3);
            }
        }

        // ---- epilogue: gather neighbour features + Linear(4->8) + ReLU ----
        const float2 o1i = p1[base + irow];
        const float vix = pi.x - o1i.x;
        const float viy = pi.y - o1i.y;

        float Wv[EMBD * 4];
#pragma unroll
        for (int t = 0; t < EMBD * 4; ++t) Wv[t] = W[t];
        float bv[EMBD];
#pragma unroll
        for (int t = 0; t < EMBD; ++t) bv[t] = bias[t];

        const int js[NKNN] = { j0, j1, j2, j3 };
        float4* orow = (float4*)(out + (size_t)(base + irow) * (NKNN * EMBD));
#pragma unroll
        for (int k = 0; k < NKNN; ++k) {
            const int j = js[k];
            const float2 pjv = p2[base + j];
            const float2 o1j = p1[base + j];
            const float f0 = pjv.x - pi.x;            // rel_pos.x
            const float f1 = pjv.y - pi.y;            // rel_pos.y
            const float f2 = (pjv.x - o1j.x) - vix;   // rel_dir.x
            const float f3 = (pjv.y - o1j.y) - viy;   // rel_dir.y
            float e[EMBD];
#pragma unroll
            for (int t = 0; t < EMBD; ++t) {
                float a = bv[t];
                a = fmaf(Wv[t * 4 + 0], f0, a);
                a = fmaf(Wv[t * 4 + 1], f1, a);
                a = fmaf(Wv[t * 4 + 2], f2, a);
                a = fmaf(Wv[t * 4 + 3], f3, a);
                e[t] = fmaxf(a, 0.0f);
            }
            orow[k * 2 + 0] = make_float4(e[0], e[1], e[2], e[3]);
            orow[k * 2 + 1] = make_float4(e[4], e[5], e[6], e[7]);
        }
    }
}

extern "C" void kernel_launch(void* const* d_in, const int* in_sizes, int n_in,
                              void* d_out, int out_size, void* d_ws, size_t ws_size,
                              hipStream_t stream) {
    const float* obs1 = (const float*)d_in[0];   // [B,N,2]
    const float* obs2 = (const float*)d_in[1];   // [B,N,2]
    const float* W    = (const float*)d_in[2];   // [8,4]
    const float* bias = (const float*)d_in[3];   // [8]
    float* out = (float*)d_out;                  // [B*N, 32]

    dim3 grid(BATCH * (NTRK / 16));              // 512 blocks
    dim3 block(32 * WAVES);                      // 4 waves each -> 2048 waves
    knn_mlp_wmma_kernel<<<grid, block, 0, stream>>>(obs1, obs2, W, bias, out);
}